// GNN_node_71159018160482
// MI455X (gfx1250) — compile-verified
//
#include <hip/hip_runtime.h>

#define N_NODES 100000
#define N_EDGES 1600000
#define DCH 64
#define NLAYER 3
#define KSEL 4096
#define BN_EPS 1e-5f

typedef float v2f __attribute__((ext_vector_type(2)));
typedef float v8f __attribute__((ext_vector_type(8)));

// ---------------------------------------------------------------------------
// Edge scatter-add: 32 lanes per edge, float2 per lane (64 channels).
// Coalesced 256B gather of h[src], atomic f32 adds into agg[dst].
// ---------------------------------------------------------------------------
__global__ __launch_bounds__(256) void gin_scatter_kernel(
    const float* __restrict__ h, const int* __restrict__ src,
    const int* __restrict__ dst, float* __restrict__ agg) {
  int t = blockIdx.x * blockDim.x + threadIdx.x;
  int e = t >> 5;
  int lane = t & 31;
  if (e >= N_EDGES) return;
  int s = src[e];
  int d = dst[e];
  const float2 v = *(const float2*)(h + (size_t)s * DCH + lane * 2);
  float* o = agg + (size_t)d * DCH + lane * 2;
  atomicAdd(o + 0, v.x);
  atomicAdd(o + 1, v.y);
}

// ---------------------------------------------------------------------------
// Fused tile GEMM: out[64rows x 64] = A @ W + bias, via V_WMMA_F32_16X16X4_F32.
// mode 1: A = inA + inB (h + agg)            [GIN pre-MLP sum]
// mode 2: A = relu(inA*scale + shift)        [BN1+ReLU applied on load]
// Also accumulates per-column sum / sum-of-squares for the following BatchNorm.
// Block = 128 threads (4 waves); wave w owns rows 16w..16w+15, all 64 columns.
// W is staged TRANSPOSED in LDS so B fragments {WsT[n][k], WsT[n][k+1]} are a
// contiguous 8B-aligned pair -> single ds_load_b64 into the even VGPR pair.
// ---------------------------------------------------------------------------
__global__ __launch_bounds__(128) void gin_gemm_kernel(
    const float* __restrict__ inA, const float* __restrict__ inB,
    const float* __restrict__ W, const float* __restrict__ bias,
    const float* __restrict__ scaleIn, const float* __restrict__ shiftIn,
    float* __restrict__ out, float* __restrict__ colsum,
    float* __restrict__ colsumsq, int mode) {
  __shared__ float Xs[64][68];   // stride 68: 16B-aligned rows, bank-stride 4
  __shared__ float WsT[64][68];  // transposed: WsT[n][k] = W[k][n]
  __shared__ float sBias[DCH], sSc[DCH], sSh[DCH];

  const int tid = threadIdx.x;
  if (tid < DCH) {
    sBias[tid] = bias[tid];
    sSc[tid] = (mode == 2) ? scaleIn[tid] : 0.0f;
    sSh[tid] = (mode == 2) ? shiftIn[tid] : 0.0f;
  }
  __syncthreads();

  const int rowBase = blockIdx.x * 64;

  // Stage W tile (64x64) cooperatively, transposing into LDS.
#pragma unroll
  for (int i = 0; i < 8; ++i) {
    int li = tid + i * 128;  // float4 index within tile
    int r = li >> 4, c4 = li & 15;
    float4 w = *(const float4*)(W + r * DCH + c4 * 4);
    WsT[c4 * 4 + 0][r] = w.x;
    WsT[c4 * 4 + 1][r] = w.y;
    WsT[c4 * 4 + 2][r] = w.z;
    WsT[c4 * 4 + 3][r] = w.w;
  }
  // Stage A tile with mode-specific elementwise fusion.
#pragma unroll
  for (int i = 0; i < 8; ++i) {
    int li = tid + i * 128;
    int r = li >> 4, c4 = li & 15;
    int g = rowBase + r;
    float4 v = make_float4(0.f, 0.f, 0.f, 0.f);
    if (g < N_NODES) {
      v = *(const float4*)(inA + (size_t)g * DCH + c4 * 4);
      if (mode == 1) {
        float4 a = *(const float4*)(inB + (size_t)g * DCH + c4 * 4);
        v.x += a.x; v.y += a.y; v.z += a.z; v.w += a.w;
      } else {
        int c = c4 * 4;
        v.x = fmaxf(0.f, fmaf(v.x, sSc[c + 0], sSh[c + 0]));
        v.y = fmaxf(0.f, fmaf(v.y, sSc[c + 1], sSh[c + 1]));
        v.z = fmaxf(0.f, fmaf(v.z, sSc[c + 2], sSh[c + 2]));
        v.w = fmaxf(0.f, fmaf(v.w, sSc[c + 3], sSh[c + 3]));
      }
    }
    *(float4*)(&Xs[r][c4 * 4]) = v;
  }
  __syncthreads();

  const int lane = tid & 31;
  const int wave = tid >> 5;
  const int m = lane & 15;          // row (A) / col (B,D) within 16-tile
  const int kb = (lane >> 4) << 1;  // K sub-offset: 0 (lanes 0-15), 2 (16-31)

  v8f acc[4];
#pragma unroll
  for (int t = 0; t < 4; ++t) acc[t] = (v8f){0.f,0.f,0.f,0.f,0.f,0.f,0.f,0.f};

  // K loop: 16 steps of K=4 -> 64 v_wmma_f32_16x16x4_f32 per wave.
#pragma unroll
  for (int k = 0; k < DCH; k += 4) {
    v2f a = *(const v2f*)(&Xs[wave * 16 + m][k + kb]);   // ds_load_b64
#pragma unroll
    for (int t = 0; t < 4; ++t) {
      v2f b = *(const v2f*)(&WsT[t * 16 + m][k + kb]);   // ds_load_b64
      acc[t] = __builtin_amdgcn_wmma_f32_16x16x4_f32(
          /*neg_a=*/false, a, /*neg_b=*/false, b,
          /*c_mod=*/(short)0, acc[t], /*reuse_a=*/false, /*reuse_b=*/false);
    }
  }

  // Bias add, store, and BN statistics accumulation.
  // C/D layout: VGPR r -> row (r + 8*(lane>=16)), col = 16t + (lane&15).
  const int halfBase = (lane >> 4) << 3;  // 0 or 8
  const int r0 = rowBase + wave * 16 + halfBase;
#pragma unroll
  for (int t = 0; t < 4; ++t) {
    const int n = t * 16 + m;
    const float bn = sBias[n];
    float cs = 0.f, cs2 = 0.f;
#pragma unroll
    for (int r = 0; r < 8; ++r) {
      int row = r0 + r;
      float v = acc[t][r] + bn;
      if (row < N_NODES) {
        out[(size_t)row * DCH + n] = v;
        cs += v;
        cs2 += v * v;
      }
    }
    atomicAdd(&colsum[n], cs);
    atomicAdd(&colsumsq[n], cs2);
  }
}

// ---------------------------------------------------------------------------
// BN prep: scale = gamma * rsqrt(var+eps), shift = beta - mean*scale
// (biased variance, matching jnp.var / PyTorch normalization).
// ---------------------------------------------------------------------------
__global__ void gin_bnprep_kernel(const float* __restrict__ colsum,
                                  const float* __restrict__ colsumsq,
                                  const float* __restrict__ gamma,
                                  const float* __restrict__ beta,
                                  float* __restrict__ scale,
                                  float* __restrict__ shift) {
  int c = threadIdx.x;
  if (c >= DCH) return;
  const float invN = 1.0f / (float)N_NODES;
  float mean = colsum[c] * invN;
  float var = colsumsq[c] * invN - mean * mean;
  float sc = gamma[c] * rsqrtf(var + BN_EPS);
  scale[c] = sc;
  shift[c] = beta[c] - mean * sc;
}

// ---------------------------------------------------------------------------
// Elementwise BN apply (+optional ReLU), float4 granularity.
// ---------------------------------------------------------------------------
__global__ __launch_bounds__(256) void gin_bnapply_kernel(
    const float* __restrict__ z, const float* __restrict__ scale,
    const float* __restrict__ shift, float* __restrict__ out, int relu) {
  int t = blockIdx.x * blockDim.x + threadIdx.x;  // float4 index
  if (t >= N_NODES * (DCH / 4)) return;
  int c = (t & 15) * 4;
  float4 v = ((const float4*)z)[t];
  v.x = fmaf(v.x, scale[c + 0], shift[c + 0]);
  v.y = fmaf(v.y, scale[c + 1], shift[c + 1]);
  v.z = fmaf(v.z, scale[c + 2], shift[c + 2]);
  v.w = fmaf(v.w, scale[c + 3], shift[c + 3]);
  if (relu) {
    v.x = fmaxf(v.x, 0.f);
    v.y = fmaxf(v.y, 0.f);
    v.z = fmaxf(v.z, 0.f);
    v.w = fmaxf(v.w, 0.f);
  }
  ((float4*)out)[t] = v;
}

// ---------------------------------------------------------------------------
// Final gather: node_select = h[batch_index].
// ---------------------------------------------------------------------------
__global__ __launch_bounds__(256) void gin_gather_kernel(
    const float* __restrict__ h, const int* __restrict__ idx,
    float* __restrict__ out) {
  int t = blockIdx.x * blockDim.x + threadIdx.x;  // float4 index
  if (t >= KSEL * (DCH / 4)) return;
  int k = t >> 4, c4 = t & 15;
  int node = idx[k];
  ((float4*)out)[t] = ((const float4*)(h + (size_t)node * DCH))[c4];
}

// ---------------------------------------------------------------------------
extern "C" void kernel_launch(void* const* d_in, const int* in_sizes, int n_in,
                              void* d_out, int out_size, void* d_ws,
                              size_t ws_size, hipStream_t stream) {
  const float* x  = (const float*)d_in[0];
  const int* edge = (const int*)d_in[1];   // [2, E]
  const int* bidx = (const int*)d_in[2];   // [K]
  const float* W1 = (const float*)d_in[3];
  const float* b1 = (const float*)d_in[4];
  const float* g1 = (const float*)d_in[5];
  const float* be1 = (const float*)d_in[6];
  const float* W2 = (const float*)d_in[7];
  const float* b2 = (const float*)d_in[8];
  const float* g2 = (const float*)d_in[9];
  const float* be2 = (const float*)d_in[10];
  float* out = (float*)d_out;

  const size_t ND = (size_t)N_NODES * DCH;
  float* ws = (float*)d_ws;
  float* agg = ws;            // [N, D]
  float* y1  = ws + ND;       // [N, D] inner-MLP activations (reused in-place)
  float* hA  = ws + 2 * ND;   // [N, D] layer output ping
  float* hB  = ws + 3 * ND;   // [N, D] layer output pong
  float* st  = ws + 4 * ND;   // stats region (512 floats)
  float* cs1 = st + 0,   *cq1 = st + 64;
  float* sc1 = st + 128, *sh1 = st + 192;
  float* cs2 = st + 256, *cq2 = st + 320;
  float* sc2 = st + 384, *sh2 = st + 448;

  const int* src = edge;
  const int* dst = edge + N_EDGES;

  const int scatterBlocks = (N_EDGES * 32) / 256;       // 200000
  const int gemmBlocks = (N_NODES + 63) / 64;           // 1563
  const int ewBlocks = (int)((ND / 4 + 255) / 256);     // 6250
  const int gatherBlocks = (KSEL * (DCH / 4) + 255) / 256;

  const float* hin = x;
  for (int l = 0; l < NLAYER; ++l) {
    hipMemsetAsync(agg, 0, ND * sizeof(float), stream);
    // zero cs1,cq1,sc1,sh1,cs2,cq2 in one shot (sc/sh overwritten by bnprep)
    hipMemsetAsync(st, 0, 384 * sizeof(float), stream);

    gin_scatter_kernel<<<scatterBlocks, 256, 0, stream>>>(hin, src, dst, agg);

    // z1 = (h + agg) @ W1 + b1, with BN1 stats
    gin_gemm_kernel<<<gemmBlocks, 128, 0, stream>>>(
        hin, agg, W1 + l * DCH * DCH, b1 + l * DCH, nullptr, nullptr, y1, cs1,
        cq1, /*mode=*/1);
    gin_bnprep_kernel<<<1, 64, 0, stream>>>(cs1, cq1, g1 + l * DCH,
                                            be1 + l * DCH, sc1, sh1);
    // z2 = relu(bn1(z1)) @ W2 + b2, with BN2 stats (in-place on y1 is safe:
    // each block reads all of its rows into LDS before any store).
    gin_gemm_kernel<<<gemmBlocks, 128, 0, stream>>>(
        y1, nullptr, W2 + l * DCH * DCH, b2 + l * DCH, sc1, sh1, y1, cs2, cq2,
        /*mode=*/2);
    gin_bnprep_kernel<<<1, 64, 0, stream>>>(cs2, cq2, g2 + l * DCH,
                                            be2 + l * DCH, sc2, sh2);

    float* hout = (l == NLAYER - 1) ? out : ((l & 1) ? hB : hA);
    gin_bnapply_kernel<<<ewBlocks, 256, 0, stream>>>(y1, sc2, sh2, hout,
                                                     (l < NLAYER - 1) ? 1 : 0);
    hin = hout;
  }

  gin_gather_kernel<<<gatherBlocks, 256, 0, stream>>>(out, bidx, out + ND);
  (void)in_sizes; (void)n_in; (void)out_size; (void)ws_size;
}